// ReLuPCA_49838800503580
// MI455X (gfx1250) — compile-verified
//
#include <hip/hip_runtime.h>
#include <math.h>

// ---------------- problem constants (x: 32 x 64 x 112 x 112 fp32) ----------
#define C_      64
#define W_      112
#define HW      12544          // 112*112
#define NIMG    32
#define MT      128            // tile columns (m values per tile)
#define TPI     98             // tiles per image = HW / MT
#define NTILES  (NIMG * TPI)   // 3136
#define MTOT_F  401408.0f
#define LDM     132            // padded LDS row stride (conflict-free col access)
#define US      68             // padded LDS row stride for U
#define ABLK    256            // gram pass blocks
#define CBLK    512            // stats / deqsum pass blocks

typedef float f32x2 __attribute__((ext_vector_type(2)));
typedef float f32x8 __attribute__((ext_vector_type(8)));
typedef int   i32x4 __attribute__((ext_vector_type(4)));

// ---------------- WMMA f32 16x16x4 ----------------------------------------
static __device__ __forceinline__ f32x8 wmma4(f32x2 a, f32x2 b, f32x8 c) {
  // 8 args: (neg_a, A, neg_b, B, c_mod, C, reuse_a, reuse_b)
  return __builtin_amdgcn_wmma_f32_16x16x4_f32(false, a, false, b, (short)0, c,
                                               false, false);
}

// A-fragment (16x4 f32): lanes 0-15 rows, VGPR0 K={0|2}, VGPR1 K={1|3}.
// B-fragment (4x16 f32): lanes 0-15 cols, VGPR0 K={0|2}, VGPR1 K={1|3}.
static __device__ __forceinline__ f32x2 wmma_frag(const float* __restrict__ p,
                                                  int rstride, int cstride,
                                                  int rbase, int cbase) {
  const int l = threadIdx.x & 31;
  const int r = rbase + (l & 15);
  const int c = cbase + ((l >> 4) << 1);
  f32x2 v;
  v.x = p[r * rstride + c * cstride];
  v.y = p[r * rstride + (c + 1) * cstride];
  return v;
}

// D (16x16 f32, 8 VGPRs): VGPR r -> M=r (lanes 0-15), M=r+8 (lanes 16-31)
static __device__ __forceinline__ void store_D(float* __restrict__ p,
                                               int rstride, int cstride,
                                               int rbase, int cbase, f32x8 d) {
  const int l = threadIdx.x & 31;
  const int cc = cbase + (l & 15);
  const int rr = rbase + ((l >> 4) << 3);
#pragma unroll
  for (int r = 0; r < 8; ++r)
    p[(rr + r) * rstride + cc * cstride] = d[r];
}

// ---------------- CDNA5 async global->LDS copy (ASYNCcnt path) -------------
static __device__ __forceinline__ void async_b128(const float* g, float* l) {
#if defined(__has_builtin) && __has_builtin(__builtin_amdgcn_global_load_async_to_lds_b128)
  __builtin_amdgcn_global_load_async_to_lds_b128(
      (__attribute__((address_space(1))) i32x4*)g,
      (__attribute__((address_space(3))) i32x4*)l, 0, 0);
#else
  const unsigned lofs =
      (unsigned)(unsigned long long)(__attribute__((address_space(3))) void*)l;
  asm volatile("global_load_async_to_lds_b128 %0, %1, off"
               :
               : "v"(lofs), "v"(g)
               : "memory");
#endif
}

static __device__ __forceinline__ void wait_async0() {
#if defined(__has_builtin) && __has_builtin(__builtin_amdgcn_s_wait_asynccnt)
  __builtin_amdgcn_s_wait_asynccnt(0);
#else
  asm volatile("s_wait_asynccnt 0x0" ::: "memory");
#endif
}

// issue async copy of one 64x128 tile (raw x) into LDS [64][LDM]
static __device__ __forceinline__ void tile_async_issue(const float* __restrict__ x,
                                                        int tile_id,
                                                        float* __restrict__ lds) {
  const int n = tile_id / TPI;
  const int hw0 = (tile_id - n * TPI) * MT;
  const float* base = x + (size_t)n * (C_ * HW) + hw0;
  const int colq = threadIdx.x & 31;  // 32 x B128 per 128-col row
  const int row0 = threadIdx.x >> 5;  // 8 rows in flight
#pragma unroll
  for (int r = row0; r < C_; r += 8)
    async_b128(base + (size_t)r * HW + colq * 4, lds + r * LDM + colq * 4);
}

// in-place relu(x) - mn on a raw tile already resident in LDS
static __device__ __forceinline__ void tile_transform(float* __restrict__ lds,
                                                      const float* __restrict__ mnl) {
  const int colq = threadIdx.x & 31;
  const int row0 = threadIdx.x >> 5;
#pragma unroll
  for (int r = row0; r < C_; r += 8) {
    float4 v = ((float4*)(lds + r * LDM))[colq];
    const float m = mnl ? mnl[r] : 0.0f;
    v.x = fmaxf(v.x, 0.0f) - m;
    v.y = fmaxf(v.y, 0.0f) - m;
    v.z = fmaxf(v.z, 0.0f) - m;
    v.w = fmaxf(v.w, 0.0f) - m;
    ((float4*)(lds + r * LDM))[colq] = v;
  }
}

// imp = U^T * tile  (K = 64 channels), result written back into tile.
// wave w owns column block w (16 cols), all 4 row blocks.
static __device__ __forceinline__ void project_tile(const float* __restrict__ uLds,
                                                    float* __restrict__ tile) {
  const int w = threadIdx.x >> 5;
  f32x8 acc[4] = {};
  for (int k = 0; k < C_; k += 4) {
    f32x2 b = wmma_frag(tile, 1, LDM, w * 16, k);   // B[k][n] = tile[k*LDM+n]
#pragma unroll
    for (int rb = 0; rb < 4; ++rb) {
      f32x2 a = wmma_frag(uLds, 1, US, rb * 16, k); // A[pr][k] = uLds[k*US+pr]
      acc[rb] = wmma4(a, b, acc[rb]);
    }
  }
  __syncthreads();
#pragma unroll
  for (int rb = 0; rb < 4; ++rb)
    store_D(tile, LDM, 1, rb * 16, w * 16, acc[rb]);
  __syncthreads();
}

// out = U * tile (tile holds deq), result written back into tile.
static __device__ __forceinline__ void backproject_tile(const float* __restrict__ uLds,
                                                        float* __restrict__ tile) {
  const int w = threadIdx.x >> 5;
  f32x8 acc[4] = {};
  for (int k = 0; k < C_; k += 4) {
    f32x2 b = wmma_frag(tile, 1, LDM, w * 16, k);
#pragma unroll
    for (int rb = 0; rb < 4; ++rb) {
      f32x2 a = wmma_frag(uLds, US, 1, rb * 16, k); // A[oc][k] = uLds[oc*US+k]
      acc[rb] = wmma4(a, b, acc[rb]);
    }
  }
  __syncthreads();
#pragma unroll
  for (int rb = 0; rb < 4; ++rb)
    store_D(tile, LDM, 1, rb * 16, w * 16, acc[rb]);
  __syncthreads();
}

static __device__ __forceinline__ float dequant(float v, float clampv,
                                                float dminv, float scalev,
                                                float eqv) {
  v = fminf(fmaxf(v, -clampv), clampv);
  if (eqv != 0.0f) return v;
  const float q = rintf((v - dminv) * scalev);  // round-half-to-even == jnp.round
  return q * (1.0f / scalev) + dminv;
}

// ===================== 1) Gram + channel sums ==============================
__global__ __launch_bounds__(256) void k_gram(const float* __restrict__ x,
                                              float* __restrict__ pG,
                                              float* __restrict__ pS) {
  __shared__ float tileA[C_ * LDM];
  __shared__ float tileB[C_ * LDM];
  __shared__ float sred[256];
  const int w = threadIdx.x >> 5;
  const int bi0 = (2 * w) >> 2, bj0 = (2 * w) & 3;
  const int bi1 = (2 * w + 1) >> 2, bj1 = (2 * w + 1) & 3;
  f32x8 acc0 = {};
  f32x8 acc1 = {};
  float ssum = 0.0f;
  const int ch = threadIdx.x >> 2, part = threadIdx.x & 3;

  tile_async_issue(x, blockIdx.x, tileA);
  int parity = 0;
  for (int t = blockIdx.x; t < NTILES; t += ABLK) {
    float* cur = parity ? tileB : tileA;
    float* nxt = parity ? tileA : tileB;
    wait_async0();
    __syncthreads();                 // all waves' async copies of `cur` landed
    tile_transform(cur, nullptr);    // relu in place
    __syncthreads();
    if (t + ABLK < NTILES) tile_async_issue(x, t + ABLK, nxt);  // overlap
    {
      const float* rp = cur + ch * LDM + part * 32;
      float s = 0.0f;
#pragma unroll
      for (int j = 0; j < 32; ++j) s += rp[j];
      ssum += s;
    }
    for (int k = 0; k < MT; k += 4) {
      f32x2 a0 = wmma_frag(cur, LDM, 1, bi0 * 16, k);
      f32x2 b0 = wmma_frag(cur, LDM, 1, bj0 * 16, k);
      acc0 = wmma4(a0, b0, acc0);
      f32x2 a1 = wmma_frag(cur, LDM, 1, bi1 * 16, k);
      f32x2 b1 = wmma_frag(cur, LDM, 1, bj1 * 16, k);
      acc1 = wmma4(a1, b1, acc1);
    }
    parity ^= 1;
  }
  sred[threadIdx.x] = ssum;
  __syncthreads();
  if (threadIdx.x < C_) {
    const float* q = sred + threadIdx.x * 4;
    pS[blockIdx.x * C_ + threadIdx.x] = (q[0] + q[1]) + (q[2] + q[3]);
  }
  float* Gp = pG + (size_t)blockIdx.x * 4096;
  store_D(Gp, 64, 1, bi0 * 16, bj0 * 16, acc0);
  store_D(Gp, 64, 1, bi1 * 16, bj1 * 16, acc1);
}

// ===================== 2) reduce Gram partials =============================
__global__ __launch_bounds__(256) void k_reduce_gram(const float* __restrict__ pG,
                                                     const float* __restrict__ pS,
                                                     float* __restrict__ G,
                                                     float* __restrict__ S) {
  const int id = blockIdx.x * 256 + threadIdx.x;
  if (id < 4096) {
    float a = 0.0f;
    for (int p = 0; p < ABLK; ++p) a += pG[(size_t)p * 4096 + id];
    G[id] = a;
  } else if (id < 4096 + C_) {
    const int c = id - 4096;
    float a = 0.0f;
    for (int p = 0; p < ABLK; ++p) a += pS[p * C_ + c];
    S[c] = a;
  }
}

// ===================== 3) cov + Jacobi eigendecomposition ==================
__global__ __launch_bounds__(64) void k_eigen(const float* __restrict__ G,
                                              const float* __restrict__ S,
                                              float* __restrict__ U,
                                              float* __restrict__ mn) {
  __shared__ float A[64 * 64];
  __shared__ float V[64 * 64];
  __shared__ float mnl[64];
  __shared__ float cs[32], sn[32];
  __shared__ int pp[32], qq[32];
  const int t = threadIdx.x;

  mnl[t] = S[t] * (1.0f / MTOT_F);
  __syncthreads();
  for (int j = 0; j < 64; ++j) {
    A[t * 64 + j] = G[t * 64 + j] * (1.0f / MTOT_F) - mnl[t] * mnl[j];
    V[t * 64 + j] = (t == j) ? 1.0f : 0.0f;
  }
  __syncthreads();

  for (int sweep = 0; sweep < 12; ++sweep) {
    for (int round = 0; round < 63; ++round) {
      if (t < 32) {  // round-robin tournament: 32 disjoint pairs
        int p, q;
        if (t == 0) { p = 63; q = round; }
        else { p = (round + t) % 63; q = (round + 63 - t) % 63; }
        if (p > q) { int tmp = p; p = q; q = tmp; }
        pp[t] = p; qq[t] = q;
        const float apq = A[p * 64 + q];
        float c = 1.0f, s = 0.0f;
        if (fabsf(apq) > 1e-12f) {
          const float theta = (A[q * 64 + q] - A[p * 64 + p]) / (2.0f * apq);
          const float tt = copysignf(1.0f, theta) /
                           (fabsf(theta) + sqrtf(1.0f + theta * theta));
          c = 1.0f / sqrtf(1.0f + tt * tt);
          s = tt * c;
        }
        cs[t] = c; sn[t] = s;
      }
      __syncthreads();
      // column rotations (thread t = row), plus eigenvector accumulation
      for (int k = 0; k < 32; ++k) {
        const int p = pp[k], q = qq[k];
        const float c = cs[k], s = sn[k];
        const float aip = A[t * 64 + p], aiq = A[t * 64 + q];
        A[t * 64 + p] = c * aip - s * aiq;
        A[t * 64 + q] = s * aip + c * aiq;
        const float vip = V[t * 64 + p], viq = V[t * 64 + q];
        V[t * 64 + p] = c * vip - s * viq;
        V[t * 64 + q] = s * vip + c * viq;
      }
      __syncthreads();
      // row rotations (thread t = column)
      for (int k = 0; k < 32; ++k) {
        const int p = pp[k], q = qq[k];
        const float c = cs[k], s = sn[k];
        const float apj = A[p * 64 + t], aqj = A[q * 64 + t];
        A[p * 64 + t] = c * apj - s * aqj;
        A[q * 64 + t] = s * apj + c * aqj;
      }
      __syncthreads();
    }
  }
  for (int j = 0; j < 64; ++j) U[t * 64 + j] = V[t * 64 + j];
  mn[t] = mnl[t];
}

// ===================== 4) project + per-channel stats ======================
__global__ __launch_bounds__(256) void k_proj_stats(const float* __restrict__ x,
                                                    const float* __restrict__ U,
                                                    const float* __restrict__ mn,
                                                    float* __restrict__ pstats) {
  __shared__ float tileA[C_ * LDM];
  __shared__ float tileB[C_ * LDM];
  __shared__ float uLds[C_ * US];
  __shared__ float mnl[C_];
  __shared__ float sred[3 * 256];
  tile_async_issue(x, blockIdx.x, tileA);
  for (int i = threadIdx.x; i < 4096; i += 256)
    uLds[(i >> 6) * US + (i & 63)] = U[i];
  if (threadIdx.x < C_) mnl[threadIdx.x] = mn[threadIdx.x];

  const int ch = threadIdx.x >> 2, part = threadIdx.x & 3;
  float sabs = 0.0f, mx = -3.402823466e38f, mnv = 3.402823466e38f;

  int parity = 0;
  for (int t = blockIdx.x; t < NTILES; t += CBLK) {
    float* cur = parity ? tileB : tileA;
    float* nxt = parity ? tileA : tileB;
    wait_async0();
    __syncthreads();
    tile_transform(cur, mnl);
    __syncthreads();
    if (t + CBLK < NTILES) tile_async_issue(x, t + CBLK, nxt);
    project_tile(uLds, cur);
    const float* rp = cur + ch * LDM + part * 32;
#pragma unroll
    for (int j = 0; j < 32; ++j) {
      const float v = rp[j];
      sabs += fabsf(v);
      mx = fmaxf(mx, v);
      mnv = fminf(mnv, v);
    }
    parity ^= 1;
  }
  sred[threadIdx.x] = sabs;
  sred[256 + threadIdx.x] = mx;
  sred[512 + threadIdx.x] = mnv;
  __syncthreads();
  if (threadIdx.x < C_) {
    const int c4 = threadIdx.x * 4;
    float* o = pstats + (size_t)blockIdx.x * 192;
    o[threadIdx.x] = (sred[c4] + sred[c4 + 1]) + (sred[c4 + 2] + sred[c4 + 3]);
    o[64 + threadIdx.x] = fmaxf(fmaxf(sred[256 + c4], sred[256 + c4 + 1]),
                                fmaxf(sred[256 + c4 + 2], sred[256 + c4 + 3]));
    o[128 + threadIdx.x] = fminf(fminf(sred[512 + c4], sred[512 + c4 + 1]),
                                 fminf(sred[512 + c4 + 2], sred[512 + c4 + 3]));
  }
}

__global__ __launch_bounds__(192) void k_reduce_stats(const float* __restrict__ pstats,
                                                      float* __restrict__ stats) {
  const int t = threadIdx.x;
  const int s = t >> 6;
  float a = pstats[t];
  for (int p = 1; p < CBLK; ++p) {
    const float v = pstats[(size_t)p * 192 + t];
    a = (s == 0) ? (a + v) : ((s == 1) ? fmaxf(a, v) : fminf(a, v));
  }
  stats[t] = a;
}

// ===================== 5) Laplace clamp + quant params =====================
__global__ __launch_bounds__(64) void k_params(const float* __restrict__ stats,
                                               float* __restrict__ params) {
  const int t = threadIdx.x;
  const float b = stats[t] * (1.0f / MTOT_F);  // mean |imp| per channel
  float clampv = 0.0f;
  if (b > 0.0f) {
    const float cc = 1.0f / 786432.0f;  // 1/(12*4^8)
    float a = 11.0f * b;
    for (int i = 0; i < 30; ++i) {
      const float e = expf(-a / b);
      const float g = cc * a - b * e;
      const float h = cc + e;
      a = a - g / h;
    }
    clampv = a;
  }
  // clip is monotone: max/min of clipped imp == clipped raw max/min
  const float dmax = fminf(fmaxf(stats[64 + t], -clampv), clampv);
  const float dmin = fminf(fmaxf(stats[128 + t], -clampv), clampv);
  const float eq = (dmax == dmin) ? 1.0f : 0.0f;
  const float scale = 254.0f / ((eq != 0.0f) ? 1.0f : (dmax - dmin));
  params[t] = clampv;
  params[64 + t] = dmin;
  params[128 + t] = scale;
  params[192 + t] = eq;
}

// ===================== 6) per-channel sum of dequantized imp ===============
__global__ __launch_bounds__(256) void k_deqsum(const float* __restrict__ x,
                                                const float* __restrict__ U,
                                                const float* __restrict__ mn,
                                                const float* __restrict__ params,
                                                float* __restrict__ pdeq) {
  __shared__ float tileA[C_ * LDM];
  __shared__ float tileB[C_ * LDM];
  __shared__ float uLds[C_ * US];
  __shared__ float mnl[C_];
  __shared__ float sred[256];
  tile_async_issue(x, blockIdx.x, tileA);
  for (int i = threadIdx.x; i < 4096; i += 256)
    uLds[(i >> 6) * US + (i & 63)] = U[i];
  if (threadIdx.x < C_) mnl[threadIdx.x] = mn[threadIdx.x];

  const int ch = threadIdx.x >> 2, part = threadIdx.x & 3;
  const float clampv = params[ch];
  const float dminv = params[64 + ch];
  const float scalev = params[128 + ch];
  const float eqv = params[192 + ch];
  float dsum = 0.0f;

  int parity = 0;
  for (int t = blockIdx.x; t < NTILES; t += CBLK) {
    float* cur = parity ? tileB : tileA;
    float* nxt = parity ? tileA : tileB;
    wait_async0();
    __syncthreads();
    tile_transform(cur, mnl);
    __syncthreads();
    if (t + CBLK < NTILES) tile_async_issue(x, t + CBLK, nxt);
    project_tile(uLds, cur);
    const float* rp = cur + ch * LDM + part * 32;
#pragma unroll
    for (int j = 0; j < 32; ++j)
      dsum += dequant(rp[j], clampv, dminv, scalev, eqv);
    parity ^= 1;
  }
  sred[threadIdx.x] = dsum;
  __syncthreads();
  if (threadIdx.x < C_) {
    const float* q = sred + threadIdx.x * 4;
    pdeq[blockIdx.x * C_ + threadIdx.x] = (q[0] + q[1]) + (q[2] + q[3]);
  }
}

__global__ __launch_bounds__(64) void k_reduce_deqsum(const float* __restrict__ pdeq,
                                                      float* __restrict__ dsum) {
  float a = 0.0f;
  for (int p = 0; p < CBLK; ++p) a += pdeq[p * C_ + threadIdx.x];
  dsum[threadIdx.x] = a;
}

// corr[c] = mn[c] - sum_k U[c][k] * mean(deq[k])
__global__ __launch_bounds__(64) void k_corr(const float* __restrict__ U,
                                             const float* __restrict__ mn,
                                             const float* __restrict__ dsum,
                                             float* __restrict__ corr) {
  const int c = threadIdx.x;
  float s = 0.0f;
  for (int k = 0; k < 64; ++k)
    s += U[c * 64 + k] * (dsum[k] * (1.0f / MTOT_F));
  corr[c] = mn[c] - s;
}

// ===================== 7) final output =====================================
__global__ __launch_bounds__(256) void k_output(const float* __restrict__ x,
                                                const float* __restrict__ U,
                                                const float* __restrict__ mn,
                                                const float* __restrict__ params,
                                                const float* __restrict__ corr,
                                                float* __restrict__ out) {
  __shared__ float tile[C_ * LDM];
  __shared__ float uLds[C_ * US];
  __shared__ float mnl[C_];
  tile_async_issue(x, blockIdx.x, tile);
  for (int i = threadIdx.x; i < 4096; i += 256)
    uLds[(i >> 6) * US + (i & 63)] = U[i];
  if (threadIdx.x < C_) mnl[threadIdx.x] = mn[threadIdx.x];

  const int ch = threadIdx.x >> 2, part = threadIdx.x & 3;
  const float clampv = params[ch];
  const float dminv = params[64 + ch];
  const float scalev = params[128 + ch];
  const float eqv = params[192 + ch];

  const int t = blockIdx.x;
  wait_async0();
  __syncthreads();
  tile_transform(tile, mnl);
  __syncthreads();
  project_tile(uLds, tile);           // tile := imp
  {
    float* rp = tile + ch * LDM + part * 32;
#pragma unroll
    for (int j = 0; j < 32; ++j)
      rp[j] = dequant(rp[j], clampv, dminv, scalev, eqv);   // tile := deq
  }
  __syncthreads();
  backproject_tile(uLds, tile);       // tile := U * deq

  const int n = t / TPI;
  const int hw0 = (t - n * TPI) * MT;
  float* base = out + (size_t)n * (C_ * HW) + hw0;
  const int colq = threadIdx.x & 31;
  const int row0 = threadIdx.x >> 5;
  for (int r = row0; r < C_; r += 8) {
    float4 v = ((const float4*)(tile + r * LDM))[colq];
    const float cr = corr[r];
    v.x += cr; v.y += cr; v.z += cr; v.w += cr;
    ((float4*)(base + (size_t)r * HW))[colq] = v;
  }
}

// ===================== launcher ============================================
extern "C" void kernel_launch(void* const* d_in, const int* in_sizes, int n_in,
                              void* d_out, int out_size, void* d_ws, size_t ws_size,
                              hipStream_t stream) {
  (void)in_sizes; (void)n_in; (void)out_size; (void)ws_size;
  const float* x = (const float*)d_in[0];
  float* out = (float*)d_out;
  float* ws = (float*)d_ws;

  // workspace layout (floats), total ~4.7 MB
  float* pG     = ws;                       // ABLK * 4096
  float* pS     = pG + (size_t)ABLK * 4096; // ABLK * 64
  float* G      = pS + (size_t)ABLK * 64;   // 4096
  float* S      = G + 4096;                 // 64
  float* U      = S + 64;                   // 4096
  float* mn     = U + 4096;                 // 64
  float* pstats = mn + 64;                  // CBLK * 192
  float* stats  = pstats + (size_t)CBLK * 192; // 192
  float* params = stats + 192;              // 256
  float* pdeq   = params + 256;             // CBLK * 64
  float* dsum   = pdeq + (size_t)CBLK * 64; // 64
  float* corr   = dsum + 64;                // 64

  k_gram<<<ABLK, 256, 0, stream>>>(x, pG, pS);
  k_reduce_gram<<<17, 256, 0, stream>>>(pG, pS, G, S);
  k_eigen<<<1, 64, 0, stream>>>(G, S, U, mn);
  k_proj_stats<<<CBLK, 256, 0, stream>>>(x, U, mn, pstats);
  k_reduce_stats<<<1, 192, 0, stream>>>(pstats, stats);
  k_params<<<1, 64, 0, stream>>>(stats, params);
  k_deqsum<<<CBLK, 256, 0, stream>>>(x, U, mn, params, pdeq);
  k_reduce_deqsum<<<1, 64, 0, stream>>>(pdeq, dsum);
  k_corr<<<1, 64, 0, stream>>>(U, mn, dsum, corr);
  k_output<<<NTILES, 256, 0, stream>>>(x, U, mn, params, corr, out);
}